// LCNLinear_59356448030848
// MI455X (gfx1250) — compile-verified
//
#include <hip/hip_runtime.h>
#include <hip/hip_bf16.h>

typedef __attribute__((ext_vector_type(2))) float v2f;
typedef __attribute__((ext_vector_type(8))) float v8f;
typedef __attribute__((ext_vector_type(4))) int   v4i;
typedef __attribute__((address_space(1))) v4i* gasp;   // global (AS1) int4*
typedef __attribute__((address_space(3))) v4i* lasp;   // LDS (AS3) int4*

#define SPA 17
#define CH  256
#define BATCH 256
#define INSZ 4352               // 17*256
#define NDIAG 5                 // band -2..+2
#define WR_ELEMS (SPA * NDIAG * CH * CH)   // 5,570,560 floats (22.3 MB)
#define XR_ELEMS ((SPA + 4) * BATCH * CH)  // 1,376,256 floats (5.5 MB), 2-joint zero pad each side

#define MT 64     // block tile M (batch)
#define NT 128    // block tile N (co)
#define KT 64     // K tile (ci)
#define PAD 68    // row pitch: lane bank = 4*row + k -> conflict-free; 272B keeps 16B alignment
#define NTILES (NDIAG * (CH / KT))   // 20 pipeline stages

#if __has_builtin(__builtin_amdgcn_global_load_async_to_lds_b128)
#define HAVE_ASYNC 1
#else
#define HAVE_ASYNC 0
#endif

__device__ __forceinline__ void async_cp16(const float* g, float* l) {
#if HAVE_ASYNC
    __builtin_amdgcn_global_load_async_to_lds_b128((gasp)g, (lasp)l, 0, 0);
#else
    *(float4*)l = *(const float4*)g;
#endif
}

__device__ __forceinline__ void wait_async() {
#if HAVE_ASYNC
#if __has_builtin(__builtin_amdgcn_s_wait_asynccnt)
    __builtin_amdgcn_s_wait_asynccnt(0);
#else
    asm volatile("s_wait_asynccnt 0" ::: "memory");
#endif
#endif
}

// ---- stage 1: scatter sparse weights into Wr[jo][d][co][ci]
__global__ __launch_bounds__(256) void lcn_scatter_w(const float* __restrict__ w,
                                                     const int* __restrict__ idx,
                                                     float* __restrict__ Wr, int nnz) {
    int k = blockIdx.x * blockDim.x + threadIdx.x;
    if (k >= nnz) return;
    int p  = idx[k];
    int r  = p / INSZ;
    int c  = p - r * INSZ;
    int jo = r % SPA, co = r / SPA;
    int ji = c % SPA, ci = c / SPA;
    int d  = ji - jo + 2;                       // guaranteed 0..4 by band mask
    Wr[(((jo * NDIAG + d) * CH + co) << 8) + ci] = w[k];
}

// ---- stage 2: permute x into Xr[ji+2][b][ci], joint-padded with zeros
__global__ __launch_bounds__(256) void lcn_reorg_x(const float* __restrict__ x,
                                                   float* __restrict__ Xr) {
    int t = blockIdx.x * blockDim.x + threadIdx.x;    // over BATCH*INSZ
    if (t >= BATCH * INSZ) return;
    int b = t / INSZ, f = t - b * INSZ;
    int ji = f % SPA, ci = f / SPA;
    Xr[(((ji + 2) * BATCH + b) << 8) + ci] = x[t];
}

// issue one tile's async copies: 8 rows-of-16-float4 for B (128x64), 4 for A (64x64)
__device__ __forceinline__ void load_tile(const float* xsrc, const float* wsrc,
                                          float (*__restrict__ As)[PAD],
                                          float (*__restrict__ Bs)[PAD],
                                          int lr, int lc4) {
#pragma unroll
    for (int i = 0; i < 8; ++i) {
        int r = lr + 16 * i;
        async_cp16(wsrc + (r << 8) + lc4 * 4, &Bs[r][lc4 * 4]);
        if (i < 4)
            async_cp16(xsrc + (r << 8) + lc4 * 4, &As[r][lc4 * 4]);
    }
}

// ---- stage 3: per-output-joint banded GEMM, f32 WMMA, double-buffered async LDS staging
// grid: (jo 0..16, mtile 0..3, ntile 0..1); block: 256 threads = 8 wave32
// wave grid 2(M) x 4(N); each wave owns a 32x32 tile = 2x2 WMMA fragments
__global__ __launch_bounds__(256) void lcn_gemm(const float* __restrict__ Wr,
                                                const float* __restrict__ Xr,
                                                const float* __restrict__ bias,
                                                float* __restrict__ y) {
    __shared__ float As[2][MT][PAD];   // 2 x 17.0 KB
    __shared__ float Bs[2][NT][PAD];   // 2 x 34.0 KB  (total 102 KB of 320 KB/WGP)

    const int jo = blockIdx.x;
    const int b0 = blockIdx.y * MT;
    const int n0 = blockIdx.z * NT;

    const int t    = threadIdx.x;
    const int lane = t & 31;
    const int w    = t >> 5;
    const int wm   = w & 1;        // 2 M-waves: row offset wm*32
    const int wn   = w >> 1;       // 4 N-waves: col offset wn*32
    const int l16  = lane & 15;
    const int lh   = lane >> 4;    // half-wave -> K pair (0,1) vs (2,3)

    const int lc4 = t & 15;        // loader float4 column
    const int lr  = t >> 4;        // loader base row 0..15

    v8f c00 = {}, c01 = {}, c10 = {}, c11 = {};

    // tile tt -> (diagonal d = tt/4, K offset kt = (tt%4)*KT)
#define TILE_PTRS(TT, XS, WS)                                                     \
    {                                                                             \
        int d_  = (TT) >> 2;                                                      \
        int kt_ = ((TT) & 3) * KT;                                                \
        XS = Xr + (((jo + d_) * BATCH + b0) << 8) + kt_;                          \
        WS = Wr + (((jo * NDIAG + d_) * CH + n0) << 8) + kt_;                     \
    }

    {   // prologue: fill buffer 0
        const float *xs, *ws;
        TILE_PTRS(0, xs, ws)
        load_tile(xs, ws, As[0], Bs[0], lr, lc4);
        wait_async();
        __syncthreads();
    }

    for (int tt = 0; tt < NTILES; ++tt) {
        const int cur = tt & 1;
        if (tt + 1 < NTILES) {   // issue next tile's async copies into the other buffer
            const float *xs, *ws;
            TILE_PTRS(tt + 1, xs, ws)
            load_tile(xs, ws, As[cur ^ 1], Bs[cur ^ 1], lr, lc4);
        }
#pragma unroll
        for (int kk = 0; kk < KT; kk += 4) {
            // A fragments (16x4 f32): lane holds row m=l16, contiguous K pair at kk+2*lh
            v2f a0 = *(const v2f*)(&As[cur][wm * 32 + l16][kk + 2 * lh]);
            v2f a1 = *(const v2f*)(&As[cur][wm * 32 + 16 + l16][kk + 2 * lh]);
            // B fragments: lane holds col n=l16, same contiguous K pair
            v2f bb0 = *(const v2f*)(&Bs[cur][wn * 32 + l16][kk + 2 * lh]);
            v2f bb1 = *(const v2f*)(&Bs[cur][wn * 32 + 16 + l16][kk + 2 * lh]);
            c00 = __builtin_amdgcn_wmma_f32_16x16x4_f32(false, a0, false, bb0,
                                                        (short)0, c00, false, false);
            c01 = __builtin_amdgcn_wmma_f32_16x16x4_f32(false, a0, false, bb1,
                                                        (short)0, c01, false, false);
            c10 = __builtin_amdgcn_wmma_f32_16x16x4_f32(false, a1, false, bb0,
                                                        (short)0, c10, false, false);
            c11 = __builtin_amdgcn_wmma_f32_16x16x4_f32(false, a1, false, bb1,
                                                        (short)0, c11, false, false);
        }
        wait_async();      // our wave's async copies for tile tt+1 have landed
        __syncthreads();   // every wave's copies have landed / buffer safe to reuse
    }
#undef TILE_PTRS

    // epilogue: D layout = lane n=l16, VGPR v -> M = v + 8*lh; add bias, scatter-store (L2 coalesces)
#define EPI(CC, IM, IN)                                                          \
    {                                                                            \
        int co = n0 + wn * 32 + 16 * (IN) + l16;                                 \
        float bv = bias[co * SPA + jo];                                          \
        _Pragma("unroll")                                                        \
        for (int v = 0; v < 8; ++v) {                                            \
            int m = b0 + wm * 32 + 16 * (IM) + v + 8 * lh;                       \
            y[m * INSZ + co * SPA + jo] = CC[v] + bv;                            \
        }                                                                        \
    }
    EPI(c00, 0, 0)
    EPI(c01, 0, 1)
    EPI(c10, 1, 0)
    EPI(c11, 1, 1)
#undef EPI
}

extern "C" void kernel_launch(void* const* d_in, const int* in_sizes, int n_in,
                              void* d_out, int out_size, void* d_ws, size_t ws_size,
                              hipStream_t stream) {
    const float* x    = (const float*)d_in[0];
    const float* wgt  = (const float*)d_in[1];
    const float* bias = (const float*)d_in[2];
    const int*   idx  = (const int*)d_in[3];
    const int nnz = in_sizes[1];

    float* Wr = (float*)d_ws;
    float* Xr = Wr + WR_ELEMS;
    size_t need = (size_t)(WR_ELEMS + XR_ELEMS) * sizeof(float);

    // zero Wr (band edges stay 0 -> uniform 5-diagonal loop) and Xr joint padding
    (void)hipMemsetAsync(d_ws, 0, need, stream);

    lcn_scatter_w<<<(nnz + 255) / 256, 256, 0, stream>>>(wgt, idx, Wr, nnz);
    lcn_reorg_x<<<(BATCH * INSZ + 255) / 256, 256, 0, stream>>>(x, Xr);

    dim3 grid(SPA, BATCH / MT, CH / NT);   // 17 x 4 x 2 = 136 blocks
    lcn_gemm<<<grid, 256, 0, stream>>>(Wr, Xr, bias, (float*)d_out);
}